// TensorProduct_52879637348698
// MI455X (gfx1250) — compile-verified
//
#include <hip/hip_runtime.h>

// ---------------------------------------------------------------------------
// Segmented tensor product 'u,v,uv'  (U=32, V=8, S0=20, S1=10, S2=12, P=32)
// Memory-bound (~0.5 GB traffic, ~1 FLOP/byte, floor ~21us @ 23.3 TB/s).
// - ballot-based path->segment grouping (no serial LDS chain)
// - async global<->LDS data movement (CDNA5 ASYNCcnt path) via inline asm
// - math via V_WMMA_F32_16X16X4_F32 (full f32 precision):
//     per (sample, out-segment): Out(32x8) = A(32xK) x B(Kx8)
// - LDS-staged 3072-f32 out row -> fully coalesced 128b streaming stores
// ---------------------------------------------------------------------------

typedef __attribute__((ext_vector_type(2))) float v2f;
typedef __attribute__((ext_vector_type(8))) float v8f;

constexpr int U_ = 32;
constexpr int V_ = 8;
constexpr int S0_ = 20;
constexpr int S1_ = 10;
constexpr int S2_ = 12;
constexpr int P_ = 32;
constexpr int X0_ROW = S0_ * U_;       // 640 floats
constexpr int X1_ROW = S1_ * V_;       // 80 floats
constexpr int OUT_ROW = S2_ * U_ * V_; // 3072 floats
constexpr int WAVES_PER_BLOCK = 4;     // 128 threads, ~62 KB LDS

#if defined(__HIP_DEVICE_COMPILE__) && defined(__gfx1250__)
#define USE_ASYNC_LDS 1
#else
#define USE_ASYNC_LDS 0
#endif

#if USE_ASYNC_LDS
// GLOBAL_LOAD_ASYNC_TO_LDS_B128: vDST = per-lane LDS byte address VGPR,
// vADDR = 64-bit global address (GV mode, SADDR = off). Tracked by ASYNCcnt.
__device__ __forceinline__ void async_load_b128(const void* gaddr, void* lds) {
    unsigned ldsoff = (unsigned)(uintptr_t)lds; // low 32b of generic LDS ptr
    asm volatile("global_load_async_to_lds_b128 %0, %1, off"
                 :: "v"(ldsoff), "v"(gaddr) : "memory");
}
// GLOBAL_STORE_ASYNC_FROM_LDS_B128: vADDR first, vSRC = LDS byte address.
__device__ __forceinline__ void async_store_b128(void* gaddr, const void* lds) {
    unsigned ldsoff = (unsigned)(uintptr_t)lds;
    asm volatile("global_store_async_from_lds_b128 %1, %0, off"
                 :: "v"(ldsoff), "v"(gaddr) : "memory");
}
__device__ __forceinline__ void wait_async0() {
    asm volatile("s_wait_asynccnt 0x0" ::: "memory");
}
__device__ __forceinline__ void wait_ds0() {
    asm volatile("s_wait_dscnt 0x0" ::: "memory");
}
#endif

__global__ __launch_bounds__(32 * WAVES_PER_BLOCK)
void tp_uvuv_wmma_kernel(const float* __restrict__ x0,
                         const float* __restrict__ x1,
                         const float* __restrict__ coeffs,
                         const int* __restrict__ idx0,
                         const int* __restrict__ idx1,
                         const int* __restrict__ idx2,
                         float* __restrict__ out,
                         int n_total) {
    // ---- block-shared path grouping ----
    __shared__ int   s_cnt[S2_];
    __shared__ int   s_plist[S2_][P_];
    __shared__ int   s_idx0[P_];
    __shared__ int   s_idx1[P_];
    __shared__ float s_cb[P_];
    // ---- per-wave staging (16B aligned for b128 LDS access) ----
    __shared__ __align__(16) float s_x0[WAVES_PER_BLOCK][X0_ROW];
    __shared__ __align__(16) float s_x1[WAVES_PER_BLOCK][X1_ROW];
    __shared__ __align__(16) float s_out[WAVES_PER_BLOCK][OUT_ROW];

    const int tid = threadIdx.x;

    // Wave-parallel grouping: lane p ranks itself among same-segment paths.
    if (tid < P_) {
        const int myseg = idx2[tid];
        unsigned mymask = 0u;
        #pragma unroll
        for (int seg = 0; seg < S2_; ++seg) {
            unsigned m = (unsigned)__ballot(myseg == seg);
            if (tid == seg) s_cnt[tid] = __popc(m); // lanes 0..11 write counts
            if (myseg == seg) mymask = m;
        }
        const int rank = __popc(mymask & ((1u << tid) - 1u));
        s_plist[myseg][rank] = tid;
        s_idx0[tid] = idx0[tid];
        s_idx1[tid] = idx1[tid];
        s_cb[tid]   = coeffs[tid];
    }
    __syncthreads();

    const int wave = tid >> 5;
    const int lane = tid & 31;
    const int n = blockIdx.x * WAVES_PER_BLOCK + wave;
    if (n >= n_total) return; // wave-uniform

    float* sx0  = s_x0[wave];
    float* sx1  = s_x1[wave];
    float* sout = s_out[wave];

    // ---- stage x0 row (640 f) and x1 row (80 f) into LDS ----
    const float4* x0row = reinterpret_cast<const float4*>(x0 + (size_t)n * X0_ROW);
    const float4* x1row = reinterpret_cast<const float4*>(x1 + (size_t)n * X1_ROW);
#if USE_ASYNC_LDS
    #pragma unroll
    for (int it = 0; it < X0_ROW / 4 / 32; ++it) {
        int i = it * 32 + lane;
        async_load_b128(x0row + i, sx0 + 4 * i);
    }
    if (lane < X1_ROW / 4) {
        async_load_b128(x1row + lane, sx1 + 4 * lane);
    }
    wait_async0(); // async-loaded LDS data visible before DS reads
#else
    #pragma unroll
    for (int it = 0; it < X0_ROW / 4 / 32; ++it) {
        int i = it * 32 + lane;
        float4 t = x0row[i];
        sx0[4 * i + 0] = t.x; sx0[4 * i + 1] = t.y;
        sx0[4 * i + 2] = t.z; sx0[4 * i + 3] = t.w;
    }
    if (lane < X1_ROW / 4) {
        float4 t = x1row[lane];
        sx1[4 * lane + 0] = t.x; sx1[4 * lane + 1] = t.y;
        sx1[4 * lane + 2] = t.z; sx1[4 * lane + 3] = t.w;
    }
#endif

    const int vcol = lane & 15; // WMMA tile N index (valid output v when < 8)
    const int hi   = lane >> 4; // 0: K slots {j,j+1}; 1: K slots {j+2,j+3}

    // ---- per output segment: Out(32x8) = A(32xK) x B(Kx8) via 16x16x4 tiles ----
    for (int s = 0; s < S2_; ++s) {
        const int ks = s_cnt[s]; // wave-uniform
        #pragma unroll
        for (int half = 0; half < 2; ++half) {
            const int ubase = half * 16;
            v8f acc = {0.f, 0.f, 0.f, 0.f, 0.f, 0.f, 0.f, 0.f};
            for (int j = 0; j < ks; j += 4) { // uniform trip count
                const int k0 = j + 2 * hi;    // slot for A/B VGPR 0
                const int k1 = k0 + 1;        // slot for A/B VGPR 1
                const int p0 = (k0 < ks) ? s_plist[s][k0] : -1;
                const int p1 = (k1 < ks) ? s_plist[s][k1] : -1;
                float a0 = 0.f, a1 = 0.f, b0 = 0.f, b1 = 0.f;
                if (p0 >= 0) {
                    a0 = sx0[s_idx0[p0] * U_ + ubase + vcol];
                    if (vcol < V_) b0 = s_cb[p0] * sx1[s_idx1[p0] * V_ + vcol];
                }
                if (p1 >= 0) {
                    a1 = sx0[s_idx0[p1] * U_ + ubase + vcol];
                    if (vcol < V_) b1 = s_cb[p1] * sx1[s_idx1[p1] * V_ + vcol];
                }
                v2f a; a[0] = a0; a[1] = a1; // A 16x4: VGPR0=K{0,2}, VGPR1=K{1,3}
                v2f b; b[0] = b0; b[1] = b1; // B 4x16 mirrored layout
                acc = __builtin_amdgcn_wmma_f32_16x16x4_f32(
                    /*neg_a=*/false, a, /*neg_b=*/false, b,
                    /*c_mod=*/(short)0, acc, /*reuse_a=*/false, /*reuse_b=*/false);
            }
            // D 16x16 f32: VGPR r -> rows r (lanes 0-15) and r+8 (lanes 16-31)
            if (vcol < V_) {
                float* dst = sout + s * (U_ * V_) + (ubase + 8 * hi) * V_ + vcol;
                #pragma unroll
                for (int r = 0; r < 8; ++r) dst[r * V_] = acc[r];
            }
        }
    }

    // ---- stream staged out row to global (24 x b128 per lane) ----
    float4* orow = reinterpret_cast<float4*>(out + (size_t)n * OUT_ROW);
#if USE_ASYNC_LDS
    wait_ds0(); // ASYNC ops are unordered vs DS: acc stores must land first
    #pragma unroll
    for (int it = 0; it < OUT_ROW / 4 / 32; ++it) {
        int i = it * 32 + lane;
        async_store_b128(orow + i, sout + 4 * i);
    }
    // S_ENDPGM performs an implicit wait-idle; ASYNCcnt drains before exit.
#else
    #pragma unroll
    for (int it = 0; it < OUT_ROW / 4 / 32; ++it) {
        int i = it * 32 + lane;
        float4 t;
        t.x = sout[4 * i + 0]; t.y = sout[4 * i + 1];
        t.z = sout[4 * i + 2]; t.w = sout[4 * i + 3];
        orow[i] = t;
    }
#endif
}

extern "C" void kernel_launch(void* const* d_in, const int* in_sizes, int n_in,
                              void* d_out, int out_size, void* d_ws, size_t ws_size,
                              hipStream_t stream) {
    const float* x0     = (const float*)d_in[0];
    const float* x1     = (const float*)d_in[1];
    const float* coeffs = (const float*)d_in[2];
    const int*   idx0   = (const int*)d_in[3];
    const int*   idx1   = (const int*)d_in[4];
    const int*   idx2   = (const int*)d_in[5];
    float*       out    = (float*)d_out;

    const int n = in_sizes[0] / X0_ROW; // N = 32768
    const int blocks = (n + WAVES_PER_BLOCK - 1) / WAVES_PER_BLOCK;
    hipLaunchKernelGGL(tp_uvuv_wmma_kernel, dim3(blocks),
                       dim3(32 * WAVES_PER_BLOCK), 0, stream,
                       x0, x1, coeffs, idx0, idx1, idx2, out, n);
}